// HierarchicalMamba_44401371906500
// MI455X (gfx1250) — compile-verified
//
#include <hip/hip_runtime.h>
#include <stdint.h>

// Hierarchical Mamba, 3 layers, stride-2 pooling between layers.
//   BATCH=2, SEQ=4096, D_MODEL=256, N_STATE=64, DT_RANK=1
//
// Per layer:
//   1) dbc_gemm_kernel    : dbc = x @ W_dbc^T via V_WMMA_F32_16X16X4_F32
//   2) chunk_scan phase 0 : per-chunk local scan (h_end, prod dA) -- 32K waves
//   3) chunk_combine      : serial combine across chunks (tiny)
//   4) chunk_scan phase 1 : re-scan chunks from correct h_start, emit y,
//                           pooling fused.  dbc/x tiles staged into LDS by the
//                           Tensor Data Mover (tensor_load_to_lds + TENSORcnt).

#define ED         256
#define NS         64
#define CDBC       129
#define DBC_STRIDE 144     // 9*16 padded columns
#define NBATCH     2
#define CH         64      // chunk length (timesteps)
#define NCMAX      64      // 4096 / CH
#define EG         8       // channels per scan workgroup (8 waves)

typedef __attribute__((ext_vector_type(2))) float v2f;
typedef __attribute__((ext_vector_type(8))) float v8f;

// ---------------- TDM support (guarded; fallback = manual LDS staging) ------
#if defined(__has_builtin)
#  if __has_builtin(__builtin_amdgcn_tensor_load_to_lds) && \
      !__has_include(<hip/amd_detail/amd_gfx1250_TDM.h>)
#    define USE_TDM 1
#  endif
#endif
#ifndef USE_TDM
#  define USE_TDM 0
#endif

__device__ __forceinline__ void wait_tensor0() {
#if defined(__has_builtin) && __has_builtin(__builtin_amdgcn_s_wait_tensorcnt)
    __builtin_amdgcn_s_wait_tensorcnt(0);
#else
    asm volatile("s_wait_tensorcnt 0x0" ::: "memory");
#endif
}

#if USE_TDM
typedef unsigned int u32x4 __attribute__((ext_vector_type(4)));
typedef int          i32x4 __attribute__((ext_vector_type(4)));
typedef int          i32x8 __attribute__((ext_vector_type(8)));

// 2D tile load: rows x dim0 f32 elements, row stride stride0 (elements),
// tile = tile1 rows x tile0 elements, into LDS byte offset lds_off.
__device__ __forceinline__ void tdm_load_2d(const void* gaddr, uint32_t lds_off,
                                            uint32_t dim0, uint32_t rows,
                                            uint32_t stride0,
                                            uint32_t tile0, uint32_t tile1)
{
    const uint64_t ga = (uint64_t)(uintptr_t)gaddr;
    u32x4 g0;
    g0[0] = 1u;                                               // count=1
    g0[1] = lds_off;                                          // lds_addr
    g0[2] = (uint32_t)ga;                                     // global_addr lo
    g0[3] = (uint32_t)((ga >> 32) & 0x1FFFFFFu) | (2u << 30); // [56:32]|type=2
    i32x8 g1;
    g1[0] = (int)(2u << 16);                                  // data_size=4B
    g1[1] = (int)((dim0 & 0xFFFFu) << 16);                    // tensor_dim0 lo
    g1[2] = (int)((dim0 >> 16) | ((rows & 0xFFFFu) << 16));   // dim0 hi|dim1 lo
    g1[3] = (int)((rows >> 16) | (tile0 << 16));              // dim1 hi|tile0
    g1[4] = (int)(tile1 & 0xFFFFu);                           // tile1, tile2=0
    g1[5] = (int)stride0;                                     // dim0_stride lo
    g1[6] = 0;
    g1[7] = 0;
    const i32x4 z = {0, 0, 0, 0};
    __builtin_amdgcn_tensor_load_to_lds(g0, g1, z, z, 0);
}
#endif

// ---------------------------------------------------------------------------
// Chunked scan.  Grid = (ED/EG, NC, B); block = 8 waves = 8 channels sharing
// one (batch, chunk).  dbc tile (64x144 f32) + x slice (64x8 f32) staged into
// LDS once per block via TDM.  phase 0: local scan -> h_end, prod dA.
// phase 1: scan from h_start, emit y with fused stride-2 pooling.
// ---------------------------------------------------------------------------
__global__ __launch_bounds__(256)
void chunk_scan_kernel(const float* __restrict__ X, const float* __restrict__ dbc,
                       const float* __restrict__ A_log, const float* __restrict__ Dp,
                       const float* __restrict__ W_dt, const float* __restrict__ b_dt,
                       float* __restrict__ hend, float* __restrict__ pend,
                       const float* __restrict__ h0buf, float* __restrict__ Out,
                       int L, int NC, int phase, int doPool)
{
    __shared__ __align__(16) float sdbc[CH * DBC_STRIDE];  // 36864 B
    __shared__ __align__(16) float sx[CH * EG];            //  2048 B

    const int eg = blockIdx.x;           // channel group
    const int c  = blockIdx.y;           // chunk
    const int b  = blockIdx.z;           // batch
    const int e0 = eg * EG;
    const int w    = threadIdx.x >> 5;
    const int lane = threadIdx.x & 31;
    const int e    = e0 + w;

    const float* gdbc = dbc + ((size_t)b * L + (size_t)c * CH) * DBC_STRIDE;
    const float* gx   = X   + ((size_t)b * L + (size_t)c * CH) * ED + e0;

#if USE_TDM
    if (threadIdx.x < 32) {
        tdm_load_2d(gdbc, (uint32_t)(uintptr_t)sdbc,
                    DBC_STRIDE, CH, DBC_STRIDE, DBC_STRIDE, CH);
        tdm_load_2d(gx, (uint32_t)(uintptr_t)sx, EG, CH, ED, EG, CH);
        wait_tensor0();
    }
#else
    for (int i = threadIdx.x; i < CH * DBC_STRIDE / 4; i += 256)
        ((float4*)sdbc)[i] = ((const float4*)gdbc)[i];
    for (int i = threadIdx.x; i < CH * EG; i += 256)
        sx[i] = gx[(size_t)(i >> 3) * ED + (i & 7)];
#endif
    __syncthreads();

    const float a0  = -__expf(A_log[(size_t)e * NS + lane]);
    const float a1  = -__expf(A_log[(size_t)e * NS + lane + 32]);
    const float wdt = W_dt[e];
    const float bdt = b_dt[e];
    const float dco = Dp[e];

    const size_t sidx = (((size_t)b * ED + e) * NC + c) * NS;
    float h0 = 0.f, h1 = 0.f, p0 = 1.f, p1 = 1.f;
    if (phase) { h0 = h0buf[sidx + lane]; h1 = h0buf[sidx + 32 + lane]; }

    auto step = [&](int t) -> float {
        const float* row = sdbc + t * DBC_STRIDE;
        const float dr  = row[0];
        const float Bn0 = row[1 + lane];
        const float Bn1 = row[33 + lane];
        const float xv  = sx[t * EG + w];
        const float z     = fmaf(dr, wdt, bdt);
        const float delta = (z > 15.f) ? z : log1pf(__expf(z));
        const float dx    = delta * xv;
        const float ea0   = __expf(delta * a0);
        const float ea1   = __expf(delta * a1);
        h0 = fmaf(h0, ea0, dx * Bn0);
        h1 = fmaf(h1, ea1, dx * Bn1);
        if (!phase) { p0 *= ea0; p1 *= ea1; return 0.f; }
        const float Cn0 = row[65 + lane];
        const float Cn1 = row[97 + lane];
        float acc = fmaf(h0, Cn0, h1 * Cn1);
#pragma unroll
        for (int off = 16; off > 0; off >>= 1)
            acc += __shfl_xor(acc, off, 32);
        return fmaf(dco, xv, acc);
    };

    for (int t = 0; t < CH; t += 2) {
        const float y0 = step(t);
        const float y1 = step(t + 1);
        if (phase && lane == 0) {
            const int l = c * CH + t;
            if (doPool) {
                Out[((size_t)b * (L >> 1) + (l >> 1)) * ED + e] = 0.5f * (y0 + y1);
            } else {
                Out[((size_t)b * L + l) * ED + e]     = y0;
                Out[((size_t)b * L + l + 1) * ED + e] = y1;
            }
        }
    }

    if (!phase) {
        hend[sidx + lane]      = h0;  hend[sidx + 32 + lane] = h1;
        pend[sidx + lane]      = p0;  pend[sidx + 32 + lane] = p1;
    }
}

// ---------------------------------------------------------------------------
// GEMM: Out[m, c] = sum_k X[m,k] * W[c,k]  (c < 129, cols padded to 144)
// One wave / 16-row M tile; 9 N tiles; V_WMMA_F32_16X16X4_F32 over K.
// ---------------------------------------------------------------------------
__global__ __launch_bounds__(32)
void dbc_gemm_kernel(const float* __restrict__ X, const float* __restrict__ W,
                     float* __restrict__ Out)
{
    const int lane = threadIdx.x;
    const int half = lane >> 4;
    const int mr   = lane & 15;
    const int m0   = blockIdx.x * 16;

    v8f acc[9];
#pragma unroll
    for (int t = 0; t < 9; ++t) acc[t] = (v8f)0.f;

    const float* xrow = X + (size_t)(m0 + mr) * ED + 2 * half;

#pragma unroll 4
    for (int k0 = 0; k0 < ED / 4; ++k0) {
        v2f a = *(const v2f*)(xrow + k0 * 4);
#pragma unroll
        for (int t = 0; t < 9; ++t) {
            const int c = t * 16 + mr;
            v2f bfrag = (v2f)0.f;
            if (c < CDBC)
                bfrag = *(const v2f*)(W + (size_t)c * ED + k0 * 4 + 2 * half);
            acc[t] = __builtin_amdgcn_wmma_f32_16x16x4_f32(
                false, a, false, bfrag, (short)0, acc[t], false, false);
        }
    }

#pragma unroll
    for (int t = 0; t < 9; ++t)
#pragma unroll
        for (int v = 0; v < 8; ++v)
            Out[(size_t)(m0 + v + 8 * half) * DBC_STRIDE + t * 16 + mr] = acc[t][v];
}

// ---------------------------------------------------------------------------
// Combine across chunks: h_start[c] = h_end[c-1] + P_end[c-1] * h_start[c-1]
// ---------------------------------------------------------------------------
__global__ __launch_bounds__(32)
void chunk_combine_kernel(const float* __restrict__ hend,
                          const float* __restrict__ pend,
                          float* __restrict__ h0buf, int NC)
{
    const int ch   = blockIdx.x;     // b*ED + e
    const int lane = threadIdx.x;
    float s0 = 0.f, s1 = 0.f;
    for (int c = 0; c < NC; ++c) {
        const size_t idx = ((size_t)ch * NC + c) * NS;
        h0buf[idx + lane]      = s0;
        h0buf[idx + 32 + lane] = s1;
        s0 = fmaf(pend[idx + lane],      s0, hend[idx + lane]);
        s1 = fmaf(pend[idx + 32 + lane], s1, hend[idx + 32 + lane]);
    }
}

// ---------------------------------------------------------------------------
extern "C" void kernel_launch(void* const* d_in, const int* in_sizes, int n_in,
                              void* d_out, int out_size, void* d_ws, size_t ws_size,
                              hipStream_t stream)
{
    (void)in_sizes; (void)n_in; (void)out_size; (void)ws_size;

    const float* x     = (const float*)d_in[0];   // (2, 4096, 256)
    const float* A_log = (const float*)d_in[1];   // (3, 256, 64)
    const float* Dp    = (const float*)d_in[2];   // (3, 256)
    const float* W_dbc = (const float*)d_in[3];   // (3, 129, 256)
    const float* W_dt  = (const float*)d_in[4];   // (3, 256, 1)
    const float* b_dt  = (const float*)d_in[5];   // (3, 256)

    // Workspace layout (floats)
    float* dbc  = (float*)d_ws;                                   // 2*4096*144
    float* x1   = dbc  + (size_t)NBATCH * 4096 * DBC_STRIDE;      // 2*2048*256
    float* x2   = x1   + (size_t)NBATCH * 2048 * ED;              // 2*1024*256
    float* hend = x2   + (size_t)NBATCH * 1024 * ED;              // 2*256*64*64
    float* pend = hend + (size_t)NBATCH * ED * NCMAX * NS;
    float* h0b  = pend + (size_t)NBATCH * ED * NCMAX * NS;

    const int Ls[3] = {4096, 2048, 1024};
    const float* xin[3] = {x, x1, x2};
    float* xout[3] = {x1, x2, (float*)d_out};

    for (int i = 0; i < 3; ++i) {
        const int L  = Ls[i];
        const int NC = L / CH;
        const int pool = (i < 2) ? 1 : 0;
        const dim3 scanGrid(ED / EG, NC, NBATCH);

        dbc_gemm_kernel<<<NBATCH * L / 16, 32, 0, stream>>>(
            xin[i], W_dbc + (size_t)i * CDBC * ED, dbc);

        chunk_scan_kernel<<<scanGrid, 256, 0, stream>>>(
            xin[i], dbc,
            A_log + (size_t)i * ED * NS, Dp + (size_t)i * ED,
            W_dt + (size_t)i * ED, b_dt + (size_t)i * ED,
            hend, pend, h0b, xout[i], L, NC, /*phase=*/0, pool);

        chunk_combine_kernel<<<NBATCH * ED, 32, 0, stream>>>(hend, pend, h0b, NC);

        chunk_scan_kernel<<<scanGrid, 256, 0, stream>>>(
            xin[i], dbc,
            A_log + (size_t)i * ED * NS, Dp + (size_t)i * ED,
            W_dt + (size_t)i * ED, b_dt + (size_t)i * ED,
            hend, pend, h0b, xout[i], L, NC, /*phase=*/1, pool);
    }
}